// CANLayer_59296318489012
// MI455X (gfx1250) — compile-verified
//
#include <hip/hip_runtime.h>
#include <hip/hip_bf16.h>

// CAN layer for MI455X (gfx1250, wave32).
// softmax over the [E,1] axis is identically 1 -> attention weights == values.
// Pipeline: fused 3-way GEMM (fp32 WMMA 16x16x4) -> edge-parallel SpMM with
// f32 atomics (L2-resident gathers) -> fused sigmoid/att_layer/elu finalize.

typedef __attribute__((ext_vector_type(2))) float v2f;
typedef __attribute__((ext_vector_type(8))) float v8f;

#define EPS_F 1e-5f

// ---------------------------------------------------------------------------
// Kernel 1: XMd = x@Wd, XMu = x@Wu, acc = (1+eps)*(x@Wi), all [N,64], fp32 WMMA.
// Block = 128 threads = 4 waves; each wave owns one 16-wide output col tile
// for ALL three matrices over a 16-row tile of x. K=64 -> 16 steps of K=4.
// A-frag (16x4 f32): lane<16 -> row=lane, K=k..k+1 ; lane>=16 -> row=lane-16,
// K=k+2..k+3 (VGPR0=K even, VGPR1=K odd). B-frag mirrored on columns.
// ---------------------------------------------------------------------------
__global__ __launch_bounds__(128) void can_gemm3_wmma(
    const float* __restrict__ x,   // [N,64]
    const float* __restrict__ Wd,  // [64,64] row-major (K x Nout)
    const float* __restrict__ Wu,
    const float* __restrict__ Wi,
    float* __restrict__ xmd,       // [N,64]
    float* __restrict__ xmu,       // [N,64]
    float* __restrict__ acc,       // [N,64]  (= d_out, seeds the SpMM sum)
    int N)
{
    const int lane = threadIdx.x & 31;
    const int wave = threadIdx.x >> 5;       // 0..3 -> output col tile
    const int row0 = blockIdx.x * 16;
    const int colBase = wave * 16;

    const int half = lane >> 4;              // 0: K+0/1, 1: K+2/3
    const int l15  = lane & 15;

    // Clamp A-row for (impossible here, N%16==0) tails; keeps EXEC all-ones.
    int arow = row0 + l15;
    if (arow >= N) arow = N - 1;
    const float* xrow = x + (size_t)arow * 64 + 2 * half;
    const int bcol = colBase + l15;

    v8f cd = {}; v8f cu = {}; v8f ci = {};

    #pragma unroll
    for (int k = 0; k < 64; k += 4) {
        v2f a;
        a.x = xrow[k + 0];
        a.y = xrow[k + 1];
        const int krow = k + 2 * half;
        v2f bd, bu, bi;
        bd.x = Wd[(size_t)(krow    ) * 64 + bcol];
        bd.y = Wd[(size_t)(krow + 1) * 64 + bcol];
        bu.x = Wu[(size_t)(krow    ) * 64 + bcol];
        bu.y = Wu[(size_t)(krow + 1) * 64 + bcol];
        bi.x = Wi[(size_t)(krow    ) * 64 + bcol];
        bi.y = Wi[(size_t)(krow + 1) * 64 + bcol];
        // (neg_a, A, neg_b, B, c_mod, C, reuse_a, reuse_b)
        cd = __builtin_amdgcn_wmma_f32_16x16x4_f32(false, a, false, bd, (short)0, cd, false, false);
        cu = __builtin_amdgcn_wmma_f32_16x16x4_f32(false, a, false, bu, (short)0, cu, false, false);
        ci = __builtin_amdgcn_wmma_f32_16x16x4_f32(false, a, false, bi, (short)0, ci, false, false);
    }

    // C/D layout: VGPR r of lane -> row = r + 8*half, col = l15 (in-tile).
    #pragma unroll
    for (int r = 0; r < 8; ++r) {
        const int row = row0 + r + 8 * half;
        if (row < N) {
            const size_t off = (size_t)row * 64 + colBase + l15;
            xmd[off] = cd[r];
            xmu[off] = cu[r];
            acc[off] = (1.0f + EPS_F) * ci[r];
        }
    }
}

// ---------------------------------------------------------------------------
// Kernel 2: acc[tgt[e]] += values[e] * xm[src[e]]  (attention == 1).
// One wave per edge; lane handles 2 channels (float2 gather + 2 f32 atomics).
// xm buffers (51 MB) are L2-resident on the 192 MB L2.
// ---------------------------------------------------------------------------
__global__ __launch_bounds__(256) void can_spmm_atomic(
    const float* __restrict__ xm,    // [N,64]
    const int*   __restrict__ tgt,   // [E]  (indices[0])
    const int*   __restrict__ src,   // [E]  (indices[1])
    const float* __restrict__ vals,  // [E]
    float* __restrict__ acc,         // [N,64]
    int E)
{
    const int e = blockIdx.x * 8 + (threadIdx.x >> 5);
    if (e >= E) return;
    const int lane = threadIdx.x & 31;

    const int s = src[e];
    const int t = tgt[e];
    const float v = vals[e];

    const float2 d = ((const float2*)(xm + (size_t)s * 64))[lane];
    float* outp = acc + (size_t)t * 64 + 2 * lane;
    atomicAdd(outp + 0, d.x * v);
    atomicAdd(outp + 1, d.y * v);
}

// ---------------------------------------------------------------------------
// Kernel 3: out = sigmoid(acc); g = elu(out_row . att_layer); out *= g.
// One wave per row; wave32 xor-shuffle reduction for the 64-wide dot.
// Operates in-place on d_out.
// ---------------------------------------------------------------------------
__global__ __launch_bounds__(256) void can_finalize(
    float* __restrict__ out,          // [N,64] in-place (holds acc on entry)
    const float* __restrict__ attL,   // [64]
    int N)
{
    const int row = blockIdx.x * 8 + (threadIdx.x >> 5);
    if (row >= N) return;
    const int lane = threadIdx.x & 31;

    const float2 a = ((const float2*)attL)[lane];
    float2* p = (float2*)(out + (size_t)row * 64) + lane;
    const float2 v = *p;

    const float o0 = 1.0f / (1.0f + __expf(-v.x));
    const float o1 = 1.0f / (1.0f + __expf(-v.y));

    float partial = o0 * a.x + o1 * a.y;
    #pragma unroll
    for (int off = 16; off > 0; off >>= 1)
        partial += __shfl_xor(partial, off, 32);

    const float g = (partial > 0.0f) ? partial : (__expf(partial) - 1.0f);

    float2 r;
    r.x = o0 * g;
    r.y = o1 * g;
    *p = r;
}

// ---------------------------------------------------------------------------
extern "C" void kernel_launch(void* const* d_in, const int* in_sizes, int n_in,
                              void* d_out, int out_size, void* d_ws, size_t ws_size,
                              hipStream_t stream) {
    const float* x      = (const float*)d_in[0];   // [N,64]
    const int*   dn_idx = (const int*)  d_in[1];   // [2,E]
    const float* dn_val = (const float*)d_in[2];   // [E]
    const int*   up_idx = (const int*)  d_in[3];   // [2,E]
    const float* up_val = (const float*)d_in[4];   // [E]
    const float* Wd     = (const float*)d_in[5];   // [64,64]
    const float* Wu     = (const float*)d_in[6];
    const float* Wi     = (const float*)d_in[7];
    // d_in[8] (att_down), d_in[9] (att_up): dead — softmax over size-1 axis == 1.
    const float* attL   = (const float*)d_in[10];  // [64,1]

    const int N = in_sizes[0] / 64;
    const int E = in_sizes[2];

    float* acc = (float*)d_out;                 // [N,64] accumulator -> final out
    float* xmd = (float*)d_ws;                  // [N,64]
    float* xmu = xmd + (size_t)N * 64;          // [N,64]

    // 1) Fused triple GEMM; seeds acc with (1+eps)*x@W_id.
    can_gemm3_wmma<<<(N + 15) / 16, 128, 0, stream>>>(x, Wd, Wu, Wi, xmd, xmu, acc, N);

    // 2) Two attention-masked (==identity) SpMMs into acc.
    const int spmmBlocks = (E + 7) / 8;
    can_spmm_atomic<<<spmmBlocks, 256, 0, stream>>>(xmd, dn_idx, dn_idx + E, dn_val, acc, E);
    can_spmm_atomic<<<spmmBlocks, 256, 0, stream>>>(xmu, up_idx, up_idx + E, up_val, acc, E);

    // 3) sigmoid + att_layer elu gate, in place.
    can_finalize<<<(N + 7) / 8, 256, 0, stream>>>(acc, attL, N);
}